// CARAFE_57758720197059
// MI455X (gfx1250) — compile-verified
//
#include <hip/hip_runtime.h>
#include <hip/hip_bf16.h>

typedef __attribute__((ext_vector_type(2))) float v2f;
typedef __attribute__((ext_vector_type(8))) float v8f;
typedef unsigned int u32x4 __attribute__((ext_vector_type(4)));
typedef int i32x4 __attribute__((ext_vector_type(4)));
typedef int i32x8 __attribute__((ext_vector_type(8)));

#define CIN   128
#define CMID  64
#define MENC  100
#define HW    4096   // 64*64
#define KENC  576    // 64*9
#define H2    128
#define W2    128
#define BN_S  0.99999500003749968750f  // 1/sqrt(1+1e-5)

// enc kernel LDS geometry
#define CG_CH   16            // channels per staging group (4 groups of 16)
#define KCG     (CG_CH * 9)   // 144 K elements per group
#define YS_ROWS 10            // 8 pixel rows + 2 halo
#define YS_COLS 66            // 64 cols + 2 halo
#define YS_CH_STRIDE (YS_ROWS * YS_COLS)  // 660

// ---------------------------------------------------------------------------
// Kernel 1: comp 1x1 conv as GEMM (M=64,K=128,N=4096 per batch) + BN + SiLU.
// One wave computes a 16(M) x 64(N) strip via V_WMMA_F32_16X16X4_F32.
// ---------------------------------------------------------------------------
__global__ void __launch_bounds__(256)
carafe_comp_gemm(const float* __restrict__ X, const float* __restrict__ Wc,
                 const float* __restrict__ g, const float* __restrict__ beta,
                 float* __restrict__ Y) {
  const int lane = threadIdx.x & 31;
  const int wave = threadIdx.x >> 5;
  const int b    = blockIdx.z;
  const int m0   = blockIdx.y * 16;
  const int n0   = (blockIdx.x * 8 + wave) * 64;
  const float* Xb = X + (size_t)b * CIN * HW;

  const int lm = lane & 15;          // M (for A) / N (for B) sub-index
  const int lk = (lane >> 4) * 2;    // K pair selector

  v8f acc[4] = {};
  for (int k = 0; k < CIN; k += 4) {
    v2f a;
    a.x = Wc[(m0 + lm) * CIN + k + lk];
    a.y = Wc[(m0 + lm) * CIN + k + lk + 1];
    if (k + 4 < CIN) {  // speculative prefetch of next B rows
      __builtin_prefetch(&Xb[(k + 4 + lk) * HW + n0 + lm], 0, 0);
      __builtin_prefetch(&Xb[(k + 5 + lk) * HW + n0 + lm], 0, 0);
    }
#pragma unroll
    for (int t = 0; t < 4; ++t) {
      const int n = n0 + t * 16 + lm;
      v2f bb;
      bb.x = Xb[(k + lk) * HW + n];
      bb.y = Xb[(k + lk + 1) * HW + n];
      acc[t] = __builtin_amdgcn_wmma_f32_16x16x4_f32(
          false, a, false, bb, (short)0, acc[t], false, false);
    }
  }

  // Epilogue: BN (mean=0, var=1) + SiLU, scatter per documented C layout.
#pragma unroll
  for (int t = 0; t < 4; ++t) {
    const int n = n0 + t * 16 + lm;
#pragma unroll
    for (int v = 0; v < 8; ++v) {
      const int m = m0 + v + (lane >> 4) * 8;
      float val = acc[t][v] * (g[m] * BN_S) + beta[m];
      val = val / (1.0f + __expf(-val));   // SiLU
      Y[((size_t)b * CMID + m) * HW + n] = val;
    }
  }
}

// ---------------------------------------------------------------------------
// Kernel 2: enc 3x3 conv as im2col GEMM (M=100 pad 112, K=576, N=4096) + BN.
// Block = 8 waves, one 16-row M tile, 8 pixel rows (wave w owns row h0+w).
// A panel is DMA'd into LDS by the Tensor Data Mover (tensor_load_to_lds,
// TENSORcnt); B operand staged as a zero-padded halo slab per 16-ch group.
// The K loop is pure ds_load + v_wmma, no bounds checks needed.
// ---------------------------------------------------------------------------
__global__ void __launch_bounds__(256)
carafe_enc_gemm(const float* __restrict__ Y, const float* __restrict__ We,
                const float* __restrict__ g, const float* __restrict__ beta,
                float* __restrict__ Wo) {
  __shared__ float Ys[CG_CH * YS_CH_STRIDE];  // 10560 floats = 42.2 KB
  __shared__ float As[16 * KCG];              //  2304 floats =  9.2 KB

  const int tid  = threadIdx.x;
  const int lane = tid & 31;
  const int wave = tid >> 5;
  const int b    = blockIdx.z;
  const int m0   = blockIdx.y * 16;
  const int h0   = blockIdx.x * 8;            // first pixel row of this block
  const float* Yb = Y + (size_t)b * CMID * HW;

  const int lm = lane & 15;
  const int lk = (lane >> 4) * 2;

  v8f acc[4] = {};
  for (int cg = 0; cg < CMID / CG_CH; ++cg) {
    __syncthreads();

    // --- TDM: DMA the A panel (16 x 144, row stride 576) into LDS. ---------
    // OOB rows (M >= 100) read zero via tensor_dim1 declared from tile start.
    if (wave == 0) {
      const unsigned long long gaddr =
          (unsigned long long)(const void*)(We + (size_t)m0 * KENC + cg * KCG);
      const unsigned int laddr = (unsigned int)(unsigned long long)(void*)As;
      u32x4 g0;
      g0[0] = 1u;                               // count=1, user mode
      g0[1] = laddr;                            // lds_addr
      g0[2] = (unsigned int)gaddr;              // global_addr[31:0]
      g0[3] = (unsigned int)((gaddr >> 32) & 0x1FFFFFFu) | 0x80000000u; // type=2
      i32x8 g1;
      g1[0] = 0x20000;                          // data_size=2 (4 bytes)
      g1[1] = (KCG & 0xFFFF) << 16;             // tensor_dim0 = 144 (lo16)
      g1[2] = ((MENC - m0) & 0xFFFF) << 16;     // tensor_dim1 rows from tile
      g1[3] = (KCG & 0xFFFF) << 16;             // tile_dim0 = 144
      g1[4] = 16;                               // tile_dim1 = 16, tile_dim2=0
      g1[5] = KENC;                             // tensor_dim0_stride = 576
      g1[6] = 0;
      g1[7] = 0;
      i32x4 gz = {0, 0, 0, 0};
#if defined(__clang_major__) && __clang_major__ >= 23
      i32x8 gz8 = {0, 0, 0, 0, 0, 0, 0, 0};
      __builtin_amdgcn_tensor_load_to_lds(g0, g1, gz, gz, gz8, 0);
#else
      __builtin_amdgcn_tensor_load_to_lds(g0, g1, gz, gz, 0);
#endif
    }

    // --- Stage B slab: channels [cg*16, cg*16+16), rows h0-1..h0+8,
    //     cols -1..64, zero-filled halo. --------------------------------
    for (int i = tid; i < CG_CH * YS_CH_STRIDE; i += 256) {
      const int ci  = i / YS_CH_STRIDE;
      const int rem = i - ci * YS_CH_STRIDE;
      const int r   = rem / YS_COLS;
      const int c   = rem - r * YS_COLS;
      const int y = h0 - 1 + r;
      const int x = c - 1;
      float v = 0.0f;
      if (((unsigned)y < 64u) && ((unsigned)x < 64u))
        v = Yb[(cg * CG_CH + ci) * HW + y * 64 + x];
      Ys[i] = v;
    }

    if (wave == 0) __builtin_amdgcn_s_wait_tensorcnt(0);
    __syncthreads();

    // --- K loop entirely out of LDS. ----------------------------------
    for (int k = 0; k < KCG; k += 4) {
      v2f a;
      a.x = As[lm * KCG + k + lk];
      a.y = As[lm * KCG + k + lk + 1];
#pragma unroll
      for (int t = 0; t < 4; ++t) {
        const int col = t * 16 + lm;            // pixel column 0..63
        v2f bb;
#pragma unroll
        for (int u = 0; u < 2; ++u) {
          const int kk = k + lk + u;            // local K index 0..143
          const int ci = kk / 9;
          const int r  = kk - ci * 9;
          const int ky = r / 3;                 // 0..2 (row offset - 1)
          const int kx = r - ky * 3;            // 0..2 (col offset - 1)
          bb[u] = Ys[ci * YS_CH_STRIDE + (wave + ky) * YS_COLS + (col + kx)];
        }
        acc[t] = __builtin_amdgcn_wmma_f32_16x16x4_f32(
            false, a, false, bb, (short)0, acc[t], false, false);
      }
    }
  }

  // Epilogue: BN, guarded store for M padding rows.
  const int hrow = h0 + wave;
#pragma unroll
  for (int t = 0; t < 4; ++t) {
    const int n = hrow * 64 + t * 16 + lm;
#pragma unroll
    for (int v = 0; v < 8; ++v) {
      const int m = m0 + v + (lane >> 4) * 8;
      if (m < MENC) {
        Wo[((size_t)b * MENC + m) * HW + n] = acc[t][v] * (g[m] * BN_S) + beta[m];
      }
    }
  }
}

// ---------------------------------------------------------------------------
// Kernel 3: pixel-shuffle gather + softmax(25) + CARAFE weighted combine.
// One thread per output pixel (b, h2, w2); loops over 128 channels.
// ---------------------------------------------------------------------------
__global__ void __launch_bounds__(256)
carafe_combine(const float* __restrict__ X, const float* __restrict__ Wo,
               float* __restrict__ out) {
  const int idx = blockIdx.x * blockDim.x + threadIdx.x;
  if (idx >= 4 * H2 * W2) return;
  const int w2 = idx & (W2 - 1);
  const int h2 = (idx >> 7) & (H2 - 1);
  const int b  = idx >> 14;
  const int h = h2 >> 1, w = w2 >> 1;
  const int r1 = h2 & 1, r2 = w2 & 1;

  // Gather 25 pre-softmax weights through pixel-shuffle channel mapping.
  const float* Wb = Wo + (size_t)b * MENC * HW + h * 64 + w;
  float wgt[25];
  float mx = -3.0e38f;
#pragma unroll
  for (int k = 0; k < 25; ++k) {
    const float v = Wb[(size_t)(k * 4 + r1 * 2 + r2) * HW];
    wgt[k] = v;
    mx = fmaxf(mx, v);
  }
  float sum = 0.0f;
#pragma unroll
  for (int k = 0; k < 25; ++k) { wgt[k] = __expf(wgt[k] - mx); sum += wgt[k]; }
  const float inv = 1.0f / sum;

  // Tap coords: Xu coord = h2 - 4 + 2*i; X coord = >>1; zero weight if OOB.
  int ys[5], xs[5];
  bool yok[5], xok[5];
#pragma unroll
  for (int i = 0; i < 5; ++i) {
    const int yy = h2 - 4 + 2 * i;
    const int xx = w2 - 4 + 2 * i;
    yok[i] = ((unsigned)yy < (unsigned)H2);
    xok[i] = ((unsigned)xx < (unsigned)W2);
    ys[i] = (min(max(yy, 0), H2 - 1)) >> 1;
    xs[i] = (min(max(xx, 0), W2 - 1)) >> 1;
  }
  float wt[25];
#pragma unroll
  for (int i = 0; i < 5; ++i)
#pragma unroll
    for (int j = 0; j < 5; ++j)
      wt[i * 5 + j] = (yok[i] && xok[j]) ? wgt[i * 5 + j] * inv : 0.0f;

  const float* Xb = X + (size_t)b * CIN * HW;
  for (int c = 0; c < CIN; ++c) {
    const float* Xc = Xb + (size_t)c * HW;
    float acc = 0.0f;
#pragma unroll
    for (int i = 0; i < 5; ++i) {
      const float* row = Xc + ys[i] * 64;
#pragma unroll
      for (int j = 0; j < 5; ++j) acc = fmaf(wt[i * 5 + j], row[xs[j]], acc);
    }
    out[(((size_t)b * CIN + c) * H2 + h2) * W2 + w2] = acc;
  }
}

// ---------------------------------------------------------------------------
extern "C" void kernel_launch(void* const* d_in, const int* in_sizes, int n_in,
                              void* d_out, int out_size, void* d_ws, size_t ws_size,
                              hipStream_t stream) {
  const float* X      = (const float*)d_in[0];
  const float* comp_w = (const float*)d_in[1];
  const float* comp_g = (const float*)d_in[2];
  const float* comp_b = (const float*)d_in[3];
  const float* enc_w  = (const float*)d_in[4];
  const float* enc_g  = (const float*)d_in[5];
  const float* enc_b  = (const float*)d_in[6];
  float* out = (float*)d_out;

  float* Ymid = (float*)d_ws;                        // [4,64,64,64]  = 4 MB
  float* Wenc = Ymid + (size_t)4 * CMID * HW;        // [4,100,64,64] = 6.25 MB

  // comp: N strips 4096/64=64 -> 8 waves/block -> grid.x=8; M tiles 64/16=4.
  carafe_comp_gemm<<<dim3(8, 4, 4), dim3(256, 1, 1), 0, stream>>>(
      X, comp_w, comp_g, comp_b, Ymid);

  // enc: 8 row-groups (8 rows each) x 7 M tiles x 4 batches.
  carafe_enc_gemm<<<dim3(8, 7, 4), dim3(256, 1, 1), 0, stream>>>(
      Ymid, enc_w, enc_g, enc_b, Wenc);

  // combine: 4*128*128 = 65536 pixels.
  carafe_combine<<<dim3(256, 1, 1), dim3(256, 1, 1), 0, stream>>>(
      X, Wenc, out);
}